// CyberMoE_24867860644286
// MI455X (gfx1250) — compile-verified
//
#include <hip/hip_runtime.h>

typedef float v2f __attribute__((ext_vector_type(2)));
typedef float v8f __attribute__((ext_vector_type(8)));

#define B_SZ 8192
#define S_SZ 64
#define H_SZ 768
#define E_SZ 5
#define L_SZ 2
#define MT   32   // rows per workgroup (two 16-row WMMA tiles per wave)
#define NTW  6    // 16-wide N tiles per wave (6*16 = 96 cols; 8 waves * 96 = 768)

__global__ __launch_bounds__(256)
void moe_fused_kernel(const float* __restrict__ hs,
                      const float* __restrict__ W1,
                      const float* __restrict__ b1,
                      const float* __restrict__ W2,
                      const float* __restrict__ b2,
                      const float* __restrict__ We,
                      const float* __restrict__ be,
                      float* __restrict__ out)
{
    __shared__ float lds_g[MT * H_SZ];          // 96 KB: relu(cls@W1+b1) tile
    __shared__ float s_l2[MT][E_SZ];            // gating logits
    __shared__ float s_all[MT][E_SZ * L_SZ];    // all_logits tile
    __shared__ float s_w0[MT], s_w1[MT];
    __shared__ int   s_i0[MT], s_i1[MT];

    const int tid  = threadIdx.x;
    const int lane = tid & 31;
    const int wave = tid >> 5;
    const int m0   = blockIdx.x * MT;

    const int lo = lane & 15;   // M (for A) / N (for B,C) lane index
    const int hi = lane >> 4;   // K-pair selector

    // ---------------- Phase 1: g = relu(cls @ W1 + b1) via f32 WMMA ----------------
    // Two 16-row M tiles per wave share each loaded B tile (2x intensity on W1).
    v8f acc0[NTW] = {};
    v8f acc1[NTW] = {};

    const float* arow0 = hs + (size_t)(m0 + lo) * (S_SZ * H_SZ);       // rows m0..m0+15
    const float* arow1 = arow0 + (size_t)16 * (S_SZ * H_SZ);           // rows m0+16..m0+31
    const int nbase = wave * (NTW * 16) + lo;

    for (int k0 = 0; k0 < H_SZ; k0 += 4) {
        const int ka = k0 + 2 * hi;
        // A 16x4 layout: lanes 0-15 hold K={k0,k0+1}, lanes 16-31 hold K={k0+2,k0+3}
        v2f a0 = *(const v2f*)(arow0 + ka);
        v2f a1 = *(const v2f*)(arow1 + ka);
        #pragma unroll
        for (int t = 0; t < NTW; ++t) {
            const int n = nbase + t * 16;
            // B 4x16 layout mirrors A: lane holds B[ka][n], B[ka+1][n]
            v2f b;
            b.x = W1[(size_t)ka * H_SZ + n];
            b.y = W1[(size_t)(ka + 1) * H_SZ + n];
            acc0[t] = __builtin_amdgcn_wmma_f32_16x16x4_f32(
                false, a0, false, b, (short)0, acc0[t], false, false);
            acc1[t] = __builtin_amdgcn_wmma_f32_16x16x4_f32(
                false, a1, false, b, (short)0, acc1[t], false, false);
        }
    }

    // C/D layout: vgpr r, lanes 0-15 -> M=r, lanes 16-31 -> M=r+8; N = lane&15
    #pragma unroll
    for (int t = 0; t < NTW; ++t) {
        const int n = nbase + t * 16;
        const float bias = b1[n];
        #pragma unroll
        for (int r = 0; r < 8; ++r) {
            const int m = r + 8 * hi;
            float v0 = acc0[t][r] + bias;
            float v1 = acc1[t][r] + bias;
            lds_g[m * H_SZ + n]        = v0 > 0.0f ? v0 : 0.0f;
            lds_g[(m + 16) * H_SZ + n] = v1 > 0.0f ? v1 : 0.0f;
        }
    }
    __syncthreads();

    // ---------------- Phase 2: gating logits = g @ W2 + b2  (32 x 5) ----------------
    if (tid < MT * E_SZ) {
        const int r = tid / E_SZ, e = tid % E_SZ;
        float s = b2[e];
        #pragma unroll 4
        for (int k = 0; k < H_SZ; ++k)
            s += lds_g[r * H_SZ + k] * W2[k * E_SZ + e];
        s_l2[r][e] = s;
    }
    __syncthreads();

    // ---------------- Phase 3a: softmax + top-2 + renorm (1 thread / row) ----------------
    if (tid < MT) {
        const int r = tid, row = m0 + r;
        float l[E_SZ], p[E_SZ];
        float mx = -3.402823466e38f;
        #pragma unroll
        for (int e = 0; e < E_SZ; ++e) { l[e] = s_l2[r][e]; mx = fmaxf(mx, l[e]); }
        float sum = 0.0f;
        #pragma unroll
        for (int e = 0; e < E_SZ; ++e) { p[e] = expf(l[e] - mx); sum += p[e]; }
        const float inv = 1.0f / sum;
        #pragma unroll
        for (int e = 0; e < E_SZ; ++e) {
            p[e] *= inv;
            out[(size_t)B_SZ * L_SZ + (size_t)row * E_SZ + e] = p[e];  // gating_probs
        }
        int i0 = 0;
        #pragma unroll
        for (int e = 1; e < E_SZ; ++e) if (p[e] > p[i0]) i0 = e;       // first max
        int i1 = (i0 == 0) ? 1 : 0;
        #pragma unroll
        for (int e = 0; e < E_SZ; ++e) if (e != i0 && p[e] > p[i1]) i1 = e; // second max
        const float ps = p[i0] + p[i1];
        s_i0[r] = i0; s_i1[r] = i1;
        s_w0[r] = p[i0] / ps; s_w1[r] = p[i1] / ps;
    }

    // ---------------- Phase 3b: all_logits = cls @ We + be  (32 x 5 x 2) ----------------
    for (int w = tid; w < MT * E_SZ * L_SZ; w += 256) {
        const int r  = w / (E_SZ * L_SZ);
        const int j  = w % (E_SZ * L_SZ);
        const int e  = j / L_SZ, li = j % L_SZ;
        const float* crow = hs + (size_t)(m0 + r) * (S_SZ * H_SZ);
        float s = be[e * L_SZ + li];
        #pragma unroll 4
        for (int h = 0; h < H_SZ; ++h)
            s += crow[h] * We[(size_t)e * H_SZ * L_SZ + (size_t)h * L_SZ + li];
        s_all[r][j] = s;
    }
    __syncthreads();

    // ---------------- Phase 4: expert_logits + final_logits ----------------
    if (tid < MT) {
        const int r = tid, row = m0 + r;
        const int i0 = s_i0[r], i1 = s_i1[r];
        const float w0 = s_w0[r], w1 = s_w1[r];
        #pragma unroll
        for (int e = 0; e < E_SZ; ++e) {
            const float sel = (e == i0 || e == i1) ? 1.0f : 0.0f;
            #pragma unroll
            for (int li = 0; li < L_SZ; ++li)
                out[(size_t)B_SZ * (L_SZ + E_SZ) + (size_t)row * (E_SZ * L_SZ) + e * L_SZ + li]
                    = s_all[r][e * L_SZ + li] * sel;                   // expert_logits
        }
        #pragma unroll
        for (int li = 0; li < L_SZ; ++li)
            out[(size_t)row * L_SZ + li] =
                s_all[r][i0 * L_SZ + li] * w0 + s_all[r][i1 * L_SZ + li] * w1; // final_logits
    }
}

extern "C" void kernel_launch(void* const* d_in, const int* in_sizes, int n_in,
                              void* d_out, int out_size, void* d_ws, size_t ws_size,
                              hipStream_t stream) {
    (void)in_sizes; (void)n_in; (void)d_ws; (void)ws_size; (void)out_size;
    const float* hs = (const float*)d_in[0];
    const float* W1 = (const float*)d_in[1];
    const float* b1 = (const float*)d_in[2];
    const float* W2 = (const float*)d_in[3];
    const float* b2 = (const float*)d_in[4];
    const float* We = (const float*)d_in[5];
    const float* be = (const float*)d_in[6];
    float* out = (float*)d_out;

    dim3 grid(B_SZ / MT);   // 256 workgroups
    dim3 block(256);        // 8 wave32s
    hipLaunchKernelGGL(moe_fused_kernel, grid, block, 0, stream,
                       hs, W1, b1, W2, b2, We, be, out);
}